// GCNLinkPredictor_77747497992413
// MI455X (gfx1250) — compile-verified
//
#include <hip/hip_runtime.h>
#include <hip/hip_bf16.h>

typedef __attribute__((ext_vector_type(2))) float v2f;
typedef __attribute__((ext_vector_type(8))) float v8f;

// ---------------- degree / normalization ----------------
__global__ void k_deg_init(float* deg, int n) {
  int i = blockIdx.x * blockDim.x + threadIdx.x;
  if (i < n) deg[i] = 1.0f;                 // self-loop contributes 1 to degree
}

__global__ void k_deg_count(const int* __restrict__ ei, int E, float* deg) {
  int e = blockIdx.x * blockDim.x + threadIdx.x;
  if (e < E) atomicAdd(&deg[ei[E + e]], 1.0f);   // row 1 = dst
}

__global__ void k_disqrt(float* deg, int n) {
  int i = blockIdx.x * blockDim.x + threadIdx.x;
  if (i < n) deg[i] = rsqrtf(deg[i]);       // deg >= 1 always (self-loop)
}

// ---------------- fp32 WMMA GEMM: C[M,N] = A[M,K] @ B[K,N] (+bias, +relu) ----------------
// grid.x = ceil(M/16); blockDim.x = (N/16)*32  (one wave per 16-wide N tile)
__global__ void k_gemm_wmma(const float* __restrict__ A, const float* __restrict__ B,
                            const float* __restrict__ bias, float* __restrict__ C,
                            int M, int K, int N, int kshift, int relu) {
  __shared__ float As[16][132];             // 16 x K tile, padded (33x16B rows -> aligned, no conflicts)
  const int m0  = blockIdx.x * 16;
  const int tid = threadIdx.x;

  // ---- async-copy the 16xK A tile into LDS (CDNA5 GLOBAL_LOAD_ASYNC_TO_LDS_B128) ----
  // one 16B chunk per iteration per lane; chunks never straddle a tile row
  const int nchunks = K << 2;               // 16*K*4B / 16B
  for (int ch = tid; ch < nchunks; ch += blockDim.x) {
    int fi = ch << 2;                       // first float of this chunk
    int r  = fi >> kshift;
    int c  = fi & (K - 1);
    int rr = m0 + r; if (rr >= M) rr = M - 1;     // clamp (row OOB values unused)
    unsigned long long ga = (unsigned long long)(A + (size_t)rr * K + c);
    unsigned int       la = (unsigned int)(size_t)(&As[r][c]);   // LDS byte offset
    asm volatile("global_load_async_to_lds_b128 %0, %1, off"
                 :: "v"(la), "v"(ga) : "memory");
  }
  asm volatile("s_wait_asynccnt 0x0" ::: "memory");
  __syncthreads();

  const int lane = tid & 31;
  const int wave = tid >> 5;
  const int half = lane >> 4;               // 0: lanes 0-15, 1: lanes 16-31
  const int l16  = lane & 15;
  const int n0   = wave * 16;

  v8f acc = {};
  for (int k0 = 0; k0 < K; k0 += 4) {
    v2f a, b;
    // A 16x4 frag: VGPR0 -> K = k0 + 2*half, VGPR1 -> K = k0 + 2*half + 1
    a[0] = As[l16][k0 + 2 * half];
    a[1] = As[l16][k0 + 2 * half + 1];
    // B 4x16 frag (row = K, striped across lanes)
    b[0] = B[(size_t)(k0 + 2 * half) * N + n0 + l16];
    b[1] = B[(size_t)(k0 + 2 * half + 1) * N + n0 + l16];
    acc = __builtin_amdgcn_wmma_f32_16x16x4_f32(false, a, false, b, (short)0, acc, false, false);
  }

  const int col = n0 + l16;
  const float bv = bias ? bias[col] : 0.0f;
#pragma unroll
  for (int r = 0; r < 8; ++r) {
    int row = m0 + r + 8 * half;            // C/D layout: VGPR r -> M = r (+8 for upper lanes)
    if (row < M) {
      float v = acc[r] + bv;
      if (relu) v = fmaxf(v, 0.0f);
      C[(size_t)row * N + col] = v;
    }
  }
}

// ---------------- GCN aggregation ----------------
// out[n,c] = disqrt[n]^2 * Hw[n,c]   (self-loop term; also initializes the buffer)
__global__ void k_agg_self(const float* __restrict__ Hw, const float* __restrict__ dis,
                           float* __restrict__ out, long long total, int cshift) {
  long long idx = (long long)blockIdx.x * blockDim.x + threadIdx.x;
  if (idx < total) {
    int node = (int)(idx >> cshift);
    float w = dis[node];
    out[idx] = w * w * Hw[idx];
  }
}

// out[dst,c] += disqrt[src]*disqrt[dst] * Hw[src,c]  (one thread per edge-channel)
// C >= 32 and waves are 32-aligned -> the edge id is wave-uniform: force scalar loads.
__global__ void k_agg_edges(const int* __restrict__ ei, int E,
                            const float* __restrict__ Hw, const float* __restrict__ dis,
                            float* __restrict__ out, int cshift) {
  long long idx = (long long)blockIdx.x * blockDim.x + threadIdx.x;
  long long total = (long long)E << cshift;
  if (idx >= total) return;
  int e = __builtin_amdgcn_readfirstlane((int)(idx >> cshift));
  int c = (int)(idx & ((1 << cshift) - 1));
  int s = ei[e];
  int d = ei[E + e];
  float nrm = dis[s] * dis[d];
  atomicAdd(&out[((size_t)d << cshift) + c], nrm * Hw[((size_t)s << cshift) + c]);
}

__global__ void k_bias_act(float* __restrict__ z, const float* __restrict__ b,
                           long long total, int cmask, int relu) {
  long long idx = (long long)blockIdx.x * blockDim.x + threadIdx.x;
  if (idx < total) {
    float v = z[idx] + b[idx & cmask];
    z[idx] = relu ? fmaxf(v, 0.0f) : v;
  }
}

// ---------------- link MLP ----------------
// P[i, 0:64] = Z[s_i], P[i, 64:128] = Z[d_i]; pair id is wave-uniform (128 >= 32).
__global__ void k_pairs(const int* __restrict__ eli, int L, const float* __restrict__ Z,
                        float* __restrict__ P) {
  long long idx = (long long)blockIdx.x * blockDim.x + threadIdx.x;
  if (idx >= ((long long)L << 7)) return;
  int i = __builtin_amdgcn_readfirstlane((int)(idx >> 7));
  int c = (int)(idx & 127);
  int node = (c < 64) ? eli[i] : eli[L + i];   // c<64 uniform per wave
  P[idx] = Z[((size_t)node << 6) + (c & 63)];
}

// out[i] = dot(h[i,0:64], wl2) + bl2   (one wave per row, shfl reduction)
__global__ void k_logits(const float* __restrict__ h, const float* __restrict__ wl2,
                         const float* __restrict__ bl2, float* __restrict__ out, int L) {
  int row  = blockIdx.x * (blockDim.x >> 5) + (threadIdx.x >> 5);
  int lane = threadIdx.x & 31;
  if (row >= L) return;
  const float* hr = h + ((size_t)row << 6);
  float v = hr[lane] * wl2[lane] + hr[lane + 32] * wl2[lane + 32];
  for (int off = 16; off > 0; off >>= 1) v += __shfl_xor(v, off, 32);
  if (lane == 0) out[row] = v + bl2[0];
}

// ---------------- launcher ----------------
extern "C" void kernel_launch(void* const* d_in, const int* in_sizes, int n_in,
                              void* d_out, int out_size, void* d_ws, size_t ws_size,
                              hipStream_t stream) {
  const float* x   = (const float*)d_in[0];
  const int*   ei  = (const int*)d_in[1];
  const int*   eli = (const int*)d_in[2];
  const float* W1  = (const float*)d_in[3];
  const float* b1  = (const float*)d_in[4];
  const float* W2  = (const float*)d_in[5];
  const float* b2  = (const float*)d_in[6];
  const float* W3  = (const float*)d_in[7];
  const float* b3  = (const float*)d_in[8];
  const float* Wl1 = (const float*)d_in[9];
  const float* bl1 = (const float*)d_in[10];
  const float* Wl2 = (const float*)d_in[11];
  const float* bl2 = (const float*)d_in[12];

  const int N = in_sizes[0] / 128;   // nodes
  const int E = in_sizes[1] / 2;     // edges
  const int L = in_sizes[2] / 2;     // label edges
  float* out = (float*)d_out;

  // workspace layout (~103 MB): disqrt | bufA (N*128) | bufB (N*128)
  float* dis  = (float*)d_ws;
  float* bufA = dis + N;
  float* bufB = bufA + (size_t)N * 128;
  float* hbuf = bufB + (size_t)N * 64;   // MLP hidden lives in upper half of bufB

  auto cdiv = [](long long a, long long b) { return (int)((a + b - 1) / b); };

  // --- degree^{-1/2} ---
  k_deg_init<<<cdiv(N, 256), 256, 0, stream>>>(dis, N);
  k_deg_count<<<cdiv(E, 256), 256, 0, stream>>>(ei, E, dis);
  k_disqrt<<<cdiv(N, 256), 256, 0, stream>>>(dis, N);

  // --- layer 1: H = X@W1 ; Z1 = relu(agg(H) + b1) ---
  k_gemm_wmma<<<cdiv(N, 16), 256, 0, stream>>>(x, W1, nullptr, bufA, N, 128, 128, 7, 0);
  k_agg_self<<<cdiv((long long)N * 128, 256), 256, 0, stream>>>(bufA, dis, bufB, (long long)N * 128, 7);
  k_agg_edges<<<cdiv((long long)E * 128, 256), 256, 0, stream>>>(ei, E, bufA, dis, bufB, 7);
  k_bias_act<<<cdiv((long long)N * 128, 256), 256, 0, stream>>>(bufB, b1, (long long)N * 128, 127, 1);

  // --- layer 2: H = Z1@W2 ; Z2 = relu(agg(H) + b2) ---
  k_gemm_wmma<<<cdiv(N, 16), 128, 0, stream>>>(bufB, W2, nullptr, bufA, N, 128, 64, 7, 0);
  k_agg_self<<<cdiv((long long)N * 64, 256), 256, 0, stream>>>(bufA, dis, bufB, (long long)N * 64, 6);
  k_agg_edges<<<cdiv((long long)E * 64, 256), 256, 0, stream>>>(ei, E, bufA, dis, bufB, 6);
  k_bias_act<<<cdiv((long long)N * 64, 256), 256, 0, stream>>>(bufB, b2, (long long)N * 64, 63, 1);

  // --- layer 3: H = Z2@W3 ; Z3 = agg(H) + b3 (no relu) ---
  k_gemm_wmma<<<cdiv(N, 16), 128, 0, stream>>>(bufB, W3, nullptr, bufA, N, 64, 64, 6, 0);
  k_agg_self<<<cdiv((long long)N * 64, 256), 256, 0, stream>>>(bufA, dis, bufB, (long long)N * 64, 6);
  k_agg_edges<<<cdiv((long long)E * 64, 256), 256, 0, stream>>>(ei, E, bufA, dis, bufB, 6);
  k_bias_act<<<cdiv((long long)N * 64, 256), 256, 0, stream>>>(bufB, b3, (long long)N * 64, 63, 0);

  // --- link MLP ---
  k_pairs<<<cdiv((long long)L * 128, 256), 256, 0, stream>>>(eli, L, bufB, bufA);
  k_gemm_wmma<<<cdiv(L, 16), 128, 0, stream>>>(bufA, Wl1, bl1, hbuf, L, 128, 64, 7, 1);
  k_logits<<<cdiv(L, 8), 256, 0, stream>>>(hbuf, Wl2, bl2, out, L);
}